// SparseGCNBlock_30906584662560
// MI455X (gfx1250) — compile-verified
//
#include <hip/hip_runtime.h>
#include <hip/hip_bf16.h>

// CDNA5 / gfx1250 SparseGCNBlock:
//   res = x@Wr + br                      (WMMA f32 16x16x4, stored to d_out)
//   hs  = dinv * (x@W)                   (WMMA f32 16x16x4, dinv baked in)
//   acc[d] = sum_{edges} hs[src]         (float4 gather + global_atomic_add_f32)
//   out = relu(LN(dinv[d]*(acc+hs[d]) + b + res)*gamma + beta)   (wave32 shfl LN)

#define IN_C 256
#define OUT_C 128

typedef float v2f __attribute__((ext_vector_type(2)));
typedef float v8f __attribute__((ext_vector_type(8)));

// -------------------------------------------------------------------------
// Init: deg = 1 (self loop), acc = 0
// -------------------------------------------------------------------------
__global__ void gcn_init_kernel(float* __restrict__ deg, float* __restrict__ acc, int n) {
    long long total = (long long)n * OUT_C;
    long long stride = (long long)gridDim.x * blockDim.x;
    long long tid = (long long)blockIdx.x * blockDim.x + threadIdx.x;
    for (long long i = tid; i < total; i += stride) acc[i] = 0.0f;
    if (tid < n) deg[tid] = 1.0f;
}

// -------------------------------------------------------------------------
// Degree accumulation over dst (float counts, exact for counts < 2^24)
// -------------------------------------------------------------------------
__global__ void gcn_deg_kernel(const long long* __restrict__ ei, float* __restrict__ deg, int nE) {
    long long i = (long long)blockIdx.x * blockDim.x + threadIdx.x;
    if (i < nE) {
        int dst = (int)ei[(long long)nE + i];
        atomicAdd(&deg[dst], 1.0f);
    }
}

__global__ void gcn_dinv_kernel(float* __restrict__ deg, int n) {
    long long i = (long long)blockIdx.x * blockDim.x + threadIdx.x;
    if (i < n) deg[i] = rsqrtf(deg[i]);   // deg >= 1 always (self loop)
}

// -------------------------------------------------------------------------
// Dual GEMM via V_WMMA_F32_16X16X4_F32.
// Block = 256 threads = 8 waves; 2 M-tiles x 4 N-groups; wave = 16x32 strip
// of BOTH x@W and x@Wr. K loop: 256 in steps of 4.
//   hs  = dinv[row] * (x@W)         -> workspace
//   res = x@Wr + br                 -> d_out (consumed in-place by finalize)
// -------------------------------------------------------------------------
__global__ void gcn_gemm_kernel(const float* __restrict__ x,
                                const float* __restrict__ W,
                                const float* __restrict__ Wr,
                                const float* __restrict__ br,
                                const float* __restrict__ dinv,
                                float* __restrict__ hs,
                                float* __restrict__ res,
                                int n) {
    const int lane  = threadIdx.x & 31;
    const int wave  = threadIdx.x >> 5;
    const int mTile = wave >> 2;      // 0..1
    const int nGrp  = wave & 3;       // 0..3
    const int row0  = blockIdx.x * 32 + mTile * 16;
    const int col0  = nGrp * 32;

    const int lhalf = lane >> 4;      // 0: K=k..k+1, 1: K=k+2..k+3
    const int l15   = lane & 15;

    int aRow = row0 + l15;
    aRow = (aRow < n) ? aRow : (n - 1);          // per-lane select, EXEC untouched
    const float* __restrict__ xA = x + (long long)aRow * IN_C + lhalf * 2;

    const int bn0 = col0 + l15;                  // B-matrix column for tile 0

    v8f acc0 = {}, acc1 = {}, accr0 = {}, accr1 = {};

    for (int k = 0; k < IN_C; k += 4) {
        // A fragment (16x4 f32): contiguous float2 per lane
        v2f a;
        a.x = xA[k];
        a.y = xA[k + 1];

        const int bk = k + lhalf * 2;
        const float* __restrict__ Wp  = W  + (long long)bk * OUT_C + bn0;
        const float* __restrict__ Wrp = Wr + (long long)bk * OUT_C + bn0;

        v2f b0, b1, c0, c1;
        b0.x = Wp[0];         b0.y = Wp[OUT_C];          // K rows bk, bk+1
        b1.x = Wp[16];        b1.y = Wp[OUT_C + 16];
        c0.x = Wrp[0];        c0.y = Wrp[OUT_C];
        c1.x = Wrp[16];       c1.y = Wrp[OUT_C + 16];

        acc0  = __builtin_amdgcn_wmma_f32_16x16x4_f32(false, a, false, b0, (short)0, acc0,  false, false);
        acc1  = __builtin_amdgcn_wmma_f32_16x16x4_f32(false, a, false, b1, (short)0, acc1,  false, false);
        accr0 = __builtin_amdgcn_wmma_f32_16x16x4_f32(false, a, false, c0, (short)0, accr0, false, false);
        accr1 = __builtin_amdgcn_wmma_f32_16x16x4_f32(false, a, false, c1, (short)0, accr1, false, false);
    }

    // C/D layout: VGPR r -> M = r (lanes 0-15) / r+8 (lanes 16-31), N = lane&15
    const int outCol = col0 + l15;
    const float br0 = br[outCol];
    const float br1 = br[outCol + 16];
#pragma unroll
    for (int r = 0; r < 8; ++r) {
        const int outRow = row0 + r + lhalf * 8;
        if (outRow < n) {
            const float dv = dinv[outRow];
            const long long base = (long long)outRow * OUT_C + outCol;
            hs[base]       = dv * acc0[r];
            hs[base + 16]  = dv * acc1[r];
            res[base]      = accr0[r] + br0;
            res[base + 16] = accr1[r] + br1;
        }
    }
}

// -------------------------------------------------------------------------
// Edge scatter: one wave per edge, lane owns 4 contiguous channels.
// acc[dst] += hs[src]
// -------------------------------------------------------------------------
__global__ void gcn_scatter_kernel(const long long* __restrict__ ei,
                                   const float* __restrict__ hs,
                                   float* __restrict__ acc, int nE) {
    const int lane = threadIdx.x & 31;
    const int wave = threadIdx.x >> 5;
    const long long e = (long long)blockIdx.x * 8 + wave;
    if (e >= nE) return;
    const int src = (int)ei[e];
    const int dst = (int)ei[(long long)nE + e];
    const float4 v = *(const float4*)(hs + (long long)src * OUT_C + lane * 4);
    float* p = acc + (long long)dst * OUT_C + lane * 4;
    atomicAdd(p + 0, v.x);
    atomicAdd(p + 1, v.y);
    atomicAdd(p + 2, v.z);
    atomicAdd(p + 3, v.w);
}

// -------------------------------------------------------------------------
// Finalize: one wave per node (4 channels/lane), LN via wave32 shfl_xor.
// out = relu( LN( dinv[d]*(acc[d]+hs[d]) + b + res[d] ) * gamma + beta )
// res is read from d_out, result written back to d_out (same wave, same row).
// -------------------------------------------------------------------------
__global__ void gcn_finalize_kernel(const float* __restrict__ hs,
                                    const float* __restrict__ acc,
                                    const float* __restrict__ dinv,
                                    const float* __restrict__ b,
                                    const float* __restrict__ gamma,
                                    const float* __restrict__ beta,
                                    float* __restrict__ out, int n) {
    const int lane = threadIdx.x & 31;
    const int wave = threadIdx.x >> 5;
    const long long node = (long long)blockIdx.x * 8 + wave;
    if (node >= n) return;
    const int c = lane * 4;
    const long long base = node * OUT_C + c;

    const float4 a4  = *(const float4*)(acc + base);
    const float4 h4  = *(const float4*)(hs + base);
    const float4 r4  = *(const float4*)(out + base);
    const float4 b4  = *(const float4*)(b + c);
    const float4 g4  = *(const float4*)(gamma + c);
    const float4 be4 = *(const float4*)(beta + c);
    const float dv = dinv[node];

    float v0 = dv * (a4.x + h4.x) + b4.x + r4.x;
    float v1 = dv * (a4.y + h4.y) + b4.y + r4.y;
    float v2 = dv * (a4.z + h4.z) + b4.z + r4.z;
    float v3 = dv * (a4.w + h4.w) + b4.w + r4.w;

    float s = v0 + v1 + v2 + v3;
#pragma unroll
    for (int m = 16; m >= 1; m >>= 1) s += __shfl_xor(s, m, 32);
    const float mean = s * (1.0f / OUT_C);

    const float d0 = v0 - mean, d1 = v1 - mean, d2 = v2 - mean, d3 = v3 - mean;
    float q = d0 * d0 + d1 * d1 + d2 * d2 + d3 * d3;
#pragma unroll
    for (int m = 16; m >= 1; m >>= 1) q += __shfl_xor(q, m, 32);
    const float inv = rsqrtf(q * (1.0f / OUT_C) + 1e-5f);

    float4 o;
    o.x = fmaxf(d0 * inv * g4.x + be4.x, 0.0f);
    o.y = fmaxf(d1 * inv * g4.y + be4.y, 0.0f);
    o.z = fmaxf(d2 * inv * g4.z + be4.z, 0.0f);
    o.w = fmaxf(d3 * inv * g4.w + be4.w, 0.0f);
    *(float4*)(out + base) = o;
}

// -------------------------------------------------------------------------
// Launch
// -------------------------------------------------------------------------
extern "C" void kernel_launch(void* const* d_in, const int* in_sizes, int n_in,
                              void* d_out, int out_size, void* d_ws, size_t ws_size,
                              hipStream_t stream) {
    const float*     x     = (const float*)d_in[0];
    const long long* ei    = (const long long*)d_in[1];   // reference dtype: int64
    const float*     W     = (const float*)d_in[2];
    const float*     b     = (const float*)d_in[3];
    const float*     Wr    = (const float*)d_in[4];
    const float*     br    = (const float*)d_in[5];
    const float*     gamma = (const float*)d_in[6];
    const float*     beta  = (const float*)d_in[7];
    float*           out   = (float*)d_out;

    const int n  = in_sizes[0] / IN_C;   // 100000
    const int nE = in_sizes[1] / 2;      // 1600000

    // Workspace layout (floats): deg/dinv [n] | hs [n*128] | acc [n*128]
    float* deg = (float*)d_ws;
    float* hs  = deg + (1 << 17);                     // 512KB-aligned region
    float* acc = hs + (long long)n * OUT_C;

    gcn_init_kernel<<<2048, 256, 0, stream>>>(deg, acc, n);
    gcn_deg_kernel<<<(nE + 255) / 256, 256, 0, stream>>>(ei, deg, nE);
    gcn_dinv_kernel<<<(n + 255) / 256, 256, 0, stream>>>(deg, n);
    gcn_gemm_kernel<<<(n + 31) / 32, 256, 0, stream>>>(x, W, Wr, br, deg, hs, out, n);
    gcn_scatter_kernel<<<(nE + 7) / 8, 256, 0, stream>>>(ei, hs, acc, nE);
    gcn_finalize_kernel<<<(n + 7) / 8, 256, 0, stream>>>(hs, acc, deg, b, gamma, beta, out, n);
}